// PostProcessor_component_6734508720233
// MI455X (gfx1250) — compile-verified
//
#include <hip/hip_runtime.h>
#include <hip/hip_bf16.h>
#include <math.h>

// ---------------------------------------------------------------------------
// Detection post-processor for MI455X (gfx1250, wave32).
// Pipeline: softmax -> box decode -> per-class stable argsort -> greedy NMS
//           -> global top-100 gather.
// CDNA5-specific paths used: GLOBAL_LOAD_ASYNC_TO_LDS_B32/B128 (ASYNCcnt DMA
// into LDS) + S_WAIT_ASYNCCNT, per CDNA5 ISA. No WMMA:
// the workload has no matmul structure (IoU = min/max/clip, NMS = masked scan).
// ---------------------------------------------------------------------------

#define NBOX   1000
#define NCLS   81
#define NFG    80          // foreground classes
#define NCOMP  11
#define IMGW   1333.0f
#define IMGH   800.0f
#define SCORE_T 0.05f
#define NMS_T   0.5f
#define NDET    100
#define XFORM_CLIP 4.135166556742356f   // log(1000/16)
#define NEG_INF (-__builtin_inff())

// Low 32 bits of a generic pointer to __shared__ memory are the LDS byte
// address (LDS aperture: addr[31:0] = LDS offset) — usable by DS/async ops.
__device__ __forceinline__ unsigned lds_addr(const void* p) {
  return (unsigned)(unsigned long long)(size_t)p;
}

__device__ __forceinline__ void async_copy_b32(unsigned lds, const void* gptr) {
  unsigned long long g = (unsigned long long)(size_t)gptr;
  asm volatile("global_load_async_to_lds_b32 %0, %1, off"
               :: "v"(lds), "v"(g) : "memory");
}

__device__ __forceinline__ void async_copy_b128(unsigned lds, const void* gptr) {
  unsigned long long g = (unsigned long long)(size_t)gptr;
  asm volatile("global_load_async_to_lds_b128 %0, %1, off"
               :: "v"(lds), "v"(g) : "memory");
}

__device__ __forceinline__ void wait_asynccnt0() {
  asm volatile("s_wait_asynccnt 0x0" ::: "memory");
}

// ---------------------------------------------------------------------------
// Kernel 1: row softmaxes. One thread per proposal row.
//   scores_c[(c-1)*N + n] = softmax(class_logits[n])[c]   (class-major)
//   comp_score[n], comp_idxf[n] = max/argmax over softmax(comp_logits[n])[1:]
// ---------------------------------------------------------------------------
__global__ void k_softmax(const float* __restrict__ cls,
                          const float* __restrict__ comp,
                          float* __restrict__ scores_c,
                          float* __restrict__ comp_score,
                          float* __restrict__ comp_idxf) {
  int n = blockIdx.x * blockDim.x + threadIdx.x;
  if (n >= NBOX) return;

  const float* row = cls + (size_t)n * NCLS;
  float m = NEG_INF;
  for (int c = 0; c < NCLS; ++c) m = fmaxf(m, row[c]);
  float s = 0.0f;
  for (int c = 0; c < NCLS; ++c) s += expf(row[c] - m);
  float inv = 1.0f / s;
  for (int c = 1; c < NCLS; ++c)
    scores_c[(size_t)(c - 1) * NBOX + n] = expf(row[c] - m) * inv;

  const float* r2 = comp + (size_t)n * NCOMP;
  float m2 = NEG_INF;
  for (int k = 0; k < NCOMP; ++k) m2 = fmaxf(m2, r2[k]);
  float s2 = 0.0f;
  for (int k = 0; k < NCOMP; ++k) s2 += expf(r2[k] - m2);
  float best = NEG_INF; int bi = 1;
  for (int k = 1; k < NCOMP; ++k) {
    float v = r2[k];
    if (v > best) { best = v; bi = k; }       // first max wins (argmax semantics)
  }
  comp_score[n] = expf(best - m2) / s2;
  comp_idxf[n]  = (float)bi;
}

// ---------------------------------------------------------------------------
// Kernel 2: BoxCoder.decode + clip_to_image, class-major output.
// ---------------------------------------------------------------------------
__global__ void k_decode(const float* __restrict__ reg,
                         const float* __restrict__ prop,
                         float* __restrict__ boxes_c) {
  int idx = blockIdx.x * blockDim.x + threadIdx.x;
  if (idx >= NFG * NBOX) return;
  int c80 = idx / NBOX;
  int n   = idx - c80 * NBOX;

  float px1 = prop[n * 4 + 0], py1 = prop[n * 4 + 1];
  float px2 = prop[n * 4 + 2], py2 = prop[n * 4 + 3];
  float w  = px2 - px1 + 1.0f, h = py2 - py1 + 1.0f;
  float cx = px1 + 0.5f * w,  cy = py1 + 0.5f * h;

  const float* r = reg + (size_t)n * (NCLS * 4) + (size_t)(c80 + 1) * 4;
  float dx = r[0] * 0.1f;                       // / WX (10)
  float dy = r[1] * 0.1f;                       // / WY (10)
  float dw = fminf(r[2] * 0.2f, XFORM_CLIP);    // / WW (5)
  float dh = fminf(r[3] * 0.2f, XFORM_CLIP);    // / WH (5)

  float pcx = dx * w + cx, pcy = dy * h + cy;
  float pw  = expf(dw) * w, ph  = expf(dh) * h;

  float x1 = fminf(fmaxf(pcx - 0.5f * pw,         0.0f), IMGW - 1.0f);
  float y1 = fminf(fmaxf(pcy - 0.5f * ph,         0.0f), IMGH - 1.0f);
  float x2 = fminf(fmaxf(pcx + 0.5f * pw - 1.0f,  0.0f), IMGW - 1.0f);
  float y2 = fminf(fmaxf(pcy + 0.5f * ph - 1.0f,  0.0f), IMGH - 1.0f);

  float4 out = make_float4(x1, y1, x2, y2);
  *(float4*)(boxes_c + (size_t)idx * 4) = out;
}

// ---------------------------------------------------------------------------
// Kernel 3: per-class stable descending argsort by rank counting.
// One block per class. Scores async-staged into LDS (ASYNCcnt path).
// rank(i) = #{ j : m(s_j) > m(s_i)  or  (m(s_j) == m(s_i) and j < i) }
// with m(v) = v if v > SCORE_T else -inf  — identical to
// argsort(-where(mask, s, -inf), stable).
// ---------------------------------------------------------------------------
__global__ void k_sort(const float* __restrict__ scores_c,
                       int*   __restrict__ order,
                       float* __restrict__ sorted_score) {
  __shared__ float s[NBOX];
  const int c = blockIdx.x, tid = threadIdx.x;
  const float* src = scores_c + (size_t)c * NBOX;

  for (int i = tid; i < NBOX; i += blockDim.x)
    async_copy_b32(lds_addr(&s[i]), src + i);
  wait_asynccnt0();
  __syncthreads();

  for (int i = tid; i < NBOX; i += blockDim.x) {
    float raw = s[i];
    float si  = (raw > SCORE_T) ? raw : NEG_INF;
    int r = 0;
    for (int j = 0; j < NBOX; ++j) {
      float rj = s[j];
      float sj = (rj > SCORE_T) ? rj : NEG_INF;
      r += (int)((sj > si) | ((sj == si) & (j < i)));
    }
    order[(size_t)c * NBOX + r]        = i;
    sorted_score[(size_t)c * NBOX + r] = raw;   // unmasked value, ref semantics
  }
}

// ---------------------------------------------------------------------------
// Kernel 4: greedy NMS per class. One block per class.
// Sorted boxes gathered into LDS with per-lane B128 async copies.
// ---------------------------------------------------------------------------
__global__ void k_nms(const int*   __restrict__ order,
                      const float* __restrict__ sorted_score,
                      const float* __restrict__ boxes_c,
                      const float* __restrict__ comp_score,
                      const float* __restrict__ comp_idxf,
                      float* __restrict__ boxes_sorted,
                      float* __restrict__ comp_s_sorted,
                      float* __restrict__ comp_i_sorted,
                      float* __restrict__ final_scores) {
  __shared__ float4        bx[NBOX];     // 16 KB
  __shared__ float         area[NBOX];   // 4 KB
  __shared__ unsigned char keep[NBOX];   // 1 KB
  const int c = blockIdx.x, tid = threadIdx.x;
  const size_t base = (size_t)c * NBOX;

  for (int i = tid; i < NBOX; i += blockDim.x) {
    int o = order[base + i];
    async_copy_b128(lds_addr(&bx[i]), boxes_c + (base + (size_t)o) * 4);
    keep[i] = (sorted_score[base + i] > SCORE_T) ? 1 : 0;
    comp_s_sorted[base + i] = comp_score[o];
    comp_i_sorted[base + i] = comp_idxf[o];
  }
  wait_asynccnt0();
  __syncthreads();

  for (int i = tid; i < NBOX; i += blockDim.x) {
    float4 b = bx[i];
    area[i] = (b.z - b.x + 1.0f) * (b.w - b.y + 1.0f);
    *(float4*)(boxes_sorted + (base + i) * 4) = b;   // persist for top-k gather
  }
  __syncthreads();

  // Greedy suppression scan: identical to the reference fori_loop body
  //   sup = (iou[i] > T) & keep[i] & (idx > i);  keep &= ~sup
  for (int i = 0; i < NBOX; ++i) {
    if (keep[i]) {                       // uniform: broadcast LDS read
      float4 bi = bx[i];
      float  ai = area[i];
      for (int j = i + 1 + tid; j < NBOX; j += blockDim.x) {
        if (!keep[j]) continue;
        float4 bj = bx[j];
        float lx = fmaxf(bi.x, bj.x), ly = fmaxf(bi.y, bj.y);
        float rx = fminf(bi.z, bj.z), ry = fminf(bi.w, bj.w);
        float iw = fmaxf(rx - lx + 1.0f, 0.0f);
        float ih = fmaxf(ry - ly + 1.0f, 0.0f);
        float inter = iw * ih;
        float iou = inter / (ai + area[j] - inter);
        if (iou > NMS_T) keep[j] = 0;
      }
    }
    __syncthreads();
  }

  for (int i = tid; i < NBOX; i += blockDim.x)
    final_scores[base + i] = keep[i] ? sorted_score[base + i] : -1.0f;
}

// ---------------------------------------------------------------------------
// Kernel 5: global top-100 by iterative argmax (smallest index wins ties).
// Single block. Consumes final_scores (fully rewritten by k_nms each launch,
// so mutation here is safe across graph replays).
// Output layout (800 floats): boxes[100*4] | scores[100] | comp_s[100] |
//                             labels[100] | comp_i[100]
// ---------------------------------------------------------------------------
__global__ void k_topk(float* __restrict__ final_scores,
                       const float* __restrict__ boxes_sorted,
                       const float* __restrict__ comp_s_sorted,
                       const float* __restrict__ comp_i_sorted,
                       float* __restrict__ out) {
  __shared__ float sval[256];
  __shared__ int   sidx[256];
  const int tid = threadIdx.x;
  const int total = NFG * NBOX;

  for (int t = 0; t < NDET; ++t) {
    float best = NEG_INF; int bi = total;
    for (int i = tid; i < total; i += blockDim.x) {
      float v = final_scores[i];
      if (v > best) { best = v; bi = i; }   // ascending i => first max wins
    }
    sval[tid] = best; sidx[tid] = bi;
    __syncthreads();
    for (int off = 128; off > 0; off >>= 1) {
      if (tid < off) {
        float v2 = sval[tid + off]; int i2 = sidx[tid + off];
        if (v2 > sval[tid] || (v2 == sval[tid] && i2 < sidx[tid])) {
          sval[tid] = v2; sidx[tid] = i2;
        }
      }
      __syncthreads();
    }
    if (tid == 0) {
      int   idx = sidx[0];
      float v   = sval[0];
      out[t * 4 + 0] = boxes_sorted[(size_t)idx * 4 + 0];
      out[t * 4 + 1] = boxes_sorted[(size_t)idx * 4 + 1];
      out[t * 4 + 2] = boxes_sorted[(size_t)idx * 4 + 2];
      out[t * 4 + 3] = boxes_sorted[(size_t)idx * 4 + 3];
      out[4 * NDET + t]         = v;
      out[5 * NDET + t]         = comp_s_sorted[idx];
      out[6 * NDET + t]         = (float)(idx / NBOX + 1);
      out[7 * NDET + t]         = comp_i_sorted[idx];
      final_scores[idx] = NEG_INF;
    }
    __syncthreads();
  }
}

// ---------------------------------------------------------------------------
extern "C" void kernel_launch(void* const* d_in, const int* in_sizes, int n_in,
                              void* d_out, int out_size, void* d_ws, size_t ws_size,
                              hipStream_t stream) {
  const float* cls  = (const float*)d_in[0];  // [1000, 81]
  const float* comp = (const float*)d_in[1];  // [1000, 11]
  const float* reg  = (const float*)d_in[2];  // [1000, 324]
  const float* prop = (const float*)d_in[3];  // [1000, 4]
  float* out = (float*)d_out;                 // 800 floats
  float* ws  = (float*)d_ws;

  // Workspace slices (floats); all 16B-aligned. Total 1,122,000 floats ~4.5MB.
  float* scores_c      = ws;                  //  80,000
  float* boxes_c       = ws + 80000;          // 320,000
  float* comp_score    = ws + 400000;         //   1,000
  float* comp_idxf     = ws + 401000;         //   1,000
  int*   order         = (int*)(ws + 402000); //  80,000
  float* sorted_score  = ws + 482000;         //  80,000
  float* boxes_sorted  = ws + 562000;         // 320,000
  float* comp_s_sorted = ws + 882000;         //  80,000
  float* comp_i_sorted = ws + 962000;         //  80,000
  float* final_scores  = ws + 1042000;        //  80,000

  k_softmax<<<(NBOX + 255) / 256, 256, 0, stream>>>(
      cls, comp, scores_c, comp_score, comp_idxf);

  k_decode<<<(NFG * NBOX + 255) / 256, 256, 0, stream>>>(
      reg, prop, boxes_c);

  k_sort<<<NFG, 256, 0, stream>>>(scores_c, order, sorted_score);

  k_nms<<<NFG, 256, 0, stream>>>(order, sorted_score, boxes_c,
                                 comp_score, comp_idxf,
                                 boxes_sorted, comp_s_sorted, comp_i_sorted,
                                 final_scores);

  k_topk<<<1, 256, 0, stream>>>(final_scores, boxes_sorted,
                                comp_s_sorted, comp_i_sorted, out);
}